// SNNStep_63127429317304
// MI455X (gfx1250) — compile-verified
//
#include <hip/hip_runtime.h>
#include <cstdint>

typedef float v2f __attribute__((ext_vector_type(2)));
typedef float v8f __attribute__((ext_vector_type(8)));

#define B_  2048
#define N_  128
#define F_  16
#define NF_ (N_ * F_)   // 2048

// ---------- shared LIF helper ----------
__device__ __forceinline__ void lif1(float mem_in, float x, float beta, float thr,
                                     float& mem_out, float& spk) {
    float m = fmaf(mem_in, beta, x);
    m = fmaxf(m, -2.0f);
    spk = (m - thr) > 0.0f ? 1.0f : 0.0f;
    mem_out = m - spk * thr;
}

// ---------- K0: pack W_s [2048,16] + W_skip_gx [2048,8] -> W_pack [2048,32] ----------
__global__ void k_pack_w(const float* __restrict__ Ws, const float* __restrict__ Wg,
                         float* __restrict__ Wp) {
    int i = blockIdx.x * blockDim.x + threadIdx.x;   // 0 .. 65535
    int row = i >> 5;
    int col = i & 31;
    float v = 0.0f;
    if (col < 16)       v = Ws[row * 16 + col];
    else if (col < 24)  v = Wg[row * 8 + (col - 16)];
    Wp[i] = v;
}

// ---------- K1: encoder LIF, float4 vectorized ----------
__global__ void k_enc_lif(const float* __restrict__ cur_in,
                          const float* __restrict__ mem_in,
                          float* __restrict__ mem_out,
                          float* __restrict__ spk_out) {
    int i = (blockIdx.x * blockDim.x + threadIdx.x) * 4;
    float4 c = *(const float4*)(cur_in + i);
    float4 m = *(const float4*)(mem_in + i);
    float4 mo, so;
    lif1(m.x, c.x, 0.8f, 0.5f, mo.x, so.x);
    lif1(m.y, c.y, 0.8f, 0.5f, mo.y, so.y);
    lif1(m.z, c.z, 0.8f, 0.5f, mo.z, so.z);
    lif1(m.w, c.w, 0.8f, 0.5f, mo.w, so.w);
    *(float4*)(mem_out + i) = mo;
    *(float4*)(spk_out + i) = so;
}

// ---------- K2: gx[b] = L_norm[b] (128x128) @ spk_enc[b] (128x16), f32 WMMA ----------
// one workgroup per batch; 8 waves, wave w owns M-tile w (rows w*16 .. w*16+15)
__global__ void __launch_bounds__(256) k_gx_wmma(const float* __restrict__ L,
                                                 const float* __restrict__ spk,
                                                 float* __restrict__ gx) {
    __shared__ float lds_spk[NF_];   // 8 KB: 128x16 spk tile for this batch
    const int b    = blockIdx.x;
    const int t    = threadIdx.x;

    // cooperative load of spk tile into LDS (512 float4 = 8KB)
    {
        const float4* src = (const float4*)(spk + (size_t)b * NF_);
        float4* dst = (float4*)lds_spk;
        dst[t]       = src[t];
        dst[t + 256] = src[t + 256];
    }
    __syncthreads();

    const int wave = t >> 5;       // M-tile index 0..7
    const int lane = t & 31;
    const int half = lane >> 4;    // 0: lanes 0-15, 1: lanes 16-31
    const int r    = lane & 15;

    // A layout (32-bit, 16x4): lanes 0-15 row M=r with K={k0,k0+1},
    //                          lanes 16-31 row M=r with K={k0+2,k0+3}
    const float* Arow = L + (size_t)b * N_ * N_ + (size_t)(wave * 16 + r) * N_;

    v8f acc = {};
    #pragma unroll 8
    for (int ks = 0; ks < 32; ++ks) {
        const int k0 = ks * 4;
        v2f a = *(const v2f*)(Arow + k0 + half * 2);
        // B layout (32-bit, 4x16): VGPR0 row K=k0+2*half, VGPR1 row K=k0+2*half+1, N=r
        v2f bf;
        bf.x = lds_spk[(k0 + half * 2    ) * F_ + r];
        bf.y = lds_spk[(k0 + half * 2 + 1) * F_ + r];
        acc = __builtin_amdgcn_wmma_f32_16x16x4_f32(
            /*neg_a=*/false, a, /*neg_b=*/false, bf,
            /*c_mod=*/(short)0, acc, /*reuse_a=*/false, /*reuse_b=*/false);
    }

    // C layout: VGPR rv -> row (rv + 8*half), N = r
    float* out = gx + (size_t)b * NF_;
    #pragma unroll
    for (int rv = 0; rv < 8; ++rv)
        out[(wave * 16 + rv + half * 8) * F_ + r] = acc[rv];
}

// ---------- K3: cur[2048,32] = gx_flat[2048,2048] @ W_pack[2048,32] ----------
// 128 waves; each owns one 16-row M-tile and both 16-wide N-tiles (shared A loads)
__global__ void __launch_bounds__(256) k_cur_wmma(const float* __restrict__ gx,
                                                  const float* __restrict__ Wp,
                                                  float* __restrict__ cur) {
    const int gw   = (blockIdx.x * blockDim.x + threadIdx.x) >> 5;  // 0..127
    const int lane = threadIdx.x & 31;
    const int half = lane >> 4;
    const int r    = lane & 15;
    const int m0   = gw * 16;

    const float* Arow = gx + (size_t)(m0 + r) * NF_;

    v8f acc0 = {}, acc1 = {};
    for (int ks = 0; ks < 512; ++ks) {
        const int k0 = ks * 4;
        v2f a = *(const v2f*)(Arow + k0 + half * 2);
        const int kr = (k0 + half * 2) * 32;
        v2f b0, b1;
        b0.x = Wp[kr + r];            // N-tile 0 (cols 0-15: cur_s)
        b0.y = Wp[kr + 32 + r];
        b1.x = Wp[kr + 16 + r];       // N-tile 1 (cols 16-23: skip_gx, 24-31 pad)
        b1.y = Wp[kr + 48 + r];
        acc0 = __builtin_amdgcn_wmma_f32_16x16x4_f32(false, a, false, b0,
                                                     (short)0, acc0, false, false);
        acc1 = __builtin_amdgcn_wmma_f32_16x16x4_f32(false, a, false, b1,
                                                     (short)0, acc1, false, false);
    }

    #pragma unroll
    for (int rv = 0; rv < 8; ++rv) {
        const int row = m0 + rv + half * 8;
        cur[row * 32 + r]      = acc0[rv];
        cur[row * 32 + 16 + r] = acc1[rv];
    }
}

// ---------- K4: per-batch tail (mem_s/hr/out LIFs + tiny matvecs) ----------
__global__ void k_tail(const float* __restrict__ cur,       // ws [B][32]
                       const float* __restrict__ mem_s_in,
                       const float* __restrict__ mem_h1_in,
                       const float* __restrict__ mem_h2_in,
                       const float* __restrict__ mem_h3_in,
                       const float* __restrict__ mem_o_in,
                       const float* __restrict__ W_skip_s,  // [16,8]
                       const float* __restrict__ W_out,     // [48,8]
                       float* __restrict__ mem_s_o,  float* __restrict__ mem_h1_o,
                       float* __restrict__ mem_h2_o, float* __restrict__ mem_h3_o,
                       float* __restrict__ mem_o_o,  float* __restrict__ spk_s_o,
                       float* __restrict__ spk_hr_o, float* __restrict__ spk_out_o) {
    int b = blockIdx.x * blockDim.x + threadIdx.x;
    if (b >= B_) return;

    float spk_s[16], shr[48];
    #pragma unroll
    for (int i = 0; i < 16; ++i) {
        float mo, s;
        lif1(mem_s_in[b * 16 + i], cur[b * 32 + i], 0.85f, 0.3f, mo, s);
        mem_s_o[b * 16 + i] = mo;
        spk_s_o[b * 16 + i] = s;
        spk_s[i] = s;
    }
    #pragma unroll
    for (int i = 0; i < 16; ++i) {
        float mo, s;
        lif1(mem_h1_in[b * 16 + i], spk_s[i], 0.80f, 0.5f, mo, s);
        mem_h1_o[b * 16 + i] = mo; shr[i] = s;
        lif1(mem_h2_in[b * 16 + i], spk_s[i], 0.90f, 0.3f, mo, s);
        mem_h2_o[b * 16 + i] = mo; shr[16 + i] = s;
        lif1(mem_h3_in[b * 16 + i], spk_s[i], 0.95f, 0.2f, mo, s);
        mem_h3_o[b * 16 + i] = mo; shr[32 + i] = s;
    }
    #pragma unroll
    for (int i = 0; i < 48; ++i) spk_hr_o[b * 48 + i] = shr[i];

    #pragma unroll
    for (int j = 0; j < 8; ++j) {
        float acc = cur[b * 32 + 16 + j];                    // gx_flat @ W_skip_gx
        #pragma unroll
        for (int i = 0; i < 48; ++i) acc = fmaf(shr[i],  W_out[i * 8 + j],    acc);
        #pragma unroll
        for (int i = 0; i < 16; ++i) acc = fmaf(spk_s[i], W_skip_s[i * 8 + j], acc);
        float mo, s;
        lif1(mem_o_in[b * 8 + j], acc, 0.9f, 0.3f, mo, s);
        mem_o_o[b * 8 + j]   = mo;
        spk_out_o[b * 8 + j] = s;
    }
}

// ---------- host-side launcher ----------
extern "C" void kernel_launch(void* const* d_in, const int* in_sizes, int n_in,
                              void* d_out, int out_size, void* d_ws, size_t ws_size,
                              hipStream_t stream) {
    const float* current_in = (const float*)d_in[0];
    const float* L_norm     = (const float*)d_in[1];
    const float* mem_enc_in = (const float*)d_in[2];
    const float* mem_s_in   = (const float*)d_in[3];
    const float* mem_h1_in  = (const float*)d_in[4];
    const float* mem_h2_in  = (const float*)d_in[5];
    const float* mem_h3_in  = (const float*)d_in[6];
    const float* mem_o_in   = (const float*)d_in[7];
    const float* W_s        = (const float*)d_in[8];
    const float* W_skip_gx  = (const float*)d_in[9];
    const float* W_skip_s   = (const float*)d_in[10];
    const float* W_out      = (const float*)d_in[11];

    const size_t n_enc = (size_t)B_ * N_ * F_;   // 4,194,304

    // d_out regions, in reference return order
    float* o_mem_enc = (float*)d_out;
    float* o_mem_s   = o_mem_enc + n_enc;
    float* o_mem_h1  = o_mem_s   + (size_t)B_ * 16;
    float* o_mem_h2  = o_mem_h1  + (size_t)B_ * 16;
    float* o_mem_h3  = o_mem_h2  + (size_t)B_ * 16;
    float* o_mem_out = o_mem_h3  + (size_t)B_ * 16;
    float* o_spk_enc = o_mem_out + (size_t)B_ * 8;
    float* o_spk_s   = o_spk_enc + n_enc;
    float* o_spk_hr  = o_spk_s   + (size_t)B_ * 16;
    float* o_spk_out = o_spk_hr  + (size_t)B_ * 48;

    // workspace layout
    float* w_gx  = (float*)d_ws;                 // [B, 2048]
    float* w_wp  = w_gx + n_enc;                 // [2048, 32]
    float* w_cur = w_wp + (size_t)2048 * 32;     // [B, 32]

    // K0: pack weights
    k_pack_w<<<256, 256, 0, stream>>>(W_s, W_skip_gx, w_wp);

    // K1: encoder LIF (float4 -> 1,048,576 threads)
    k_enc_lif<<<4096, 256, 0, stream>>>(current_in, mem_enc_in, o_mem_enc, o_spk_enc);

    // K2: batched WMMA graph message passing (one block per batch)
    k_gx_wmma<<<B_, 256, 0, stream>>>(L_norm, o_spk_enc, w_gx);

    // K3: WMMA projection to cur_s / skip_gx (128 waves = 16 blocks)
    k_cur_wmma<<<16, 256, 0, stream>>>(w_gx, w_wp, w_cur);

    // K4: per-batch tail
    k_tail<<<(B_ + 255) / 256, 256, 0, stream>>>(w_cur, mem_s_in, mem_h1_in, mem_h2_in,
                                                 mem_h3_in, mem_o_in, W_skip_s, W_out,
                                                 o_mem_s, o_mem_h1, o_mem_h2, o_mem_h3,
                                                 o_mem_out, o_spk_s, o_spk_hr, o_spk_out);
}